// LocalGrouper_84404697301378
// MI455X (gfx1250) — compile-verified
//
#include <hip/hip_runtime.h>
#include <math.h>

// Problem constants (match reference setup_inputs)
constexpr int kB = 8;
constexpr int kN = 4096;
constexpr int kC = 64;
constexpr int kS = 1024;   // NUM_SAMPLES
constexpr int kK = 32;     // K_NEIGHBORS
constexpr float kEPS = 1e-5f;

typedef float v2f __attribute__((ext_vector_type(2)));
typedef float v8f __attribute__((ext_vector_type(8)));

// ---------------------------------------------------------------------------
// Kernel 1: furthest point sampling. One workgroup (1024 thr = 32 waves) per
// batch. Each thread owns 4 points in registers; per iteration: update min
// distance, wave32 shuffle argmax (first-index tie-break), cross-wave LDS
// reduction, broadcast winner. Only two barriers per iteration.
// ---------------------------------------------------------------------------
__global__ __launch_bounds__(1024) void fps_kernel(const float* __restrict__ xyz,
                                                   int* __restrict__ fps_idx)
{
    const int b = blockIdx.x;
    const int t = threadIdx.x;
    const float* X = xyz + (size_t)b * kN * 3;

    float px[4], py[4], pz[4], md[4];
#pragma unroll
    for (int j = 0; j < 4; ++j) {
        int p = t * 4 + j;
        px[j] = X[p * 3 + 0];
        py[j] = X[p * 3 + 1];
        pz[j] = X[p * 3 + 2];
        md[j] = 1e10f;
    }

    __shared__ float red_v[32];
    __shared__ int   red_i[32];
    __shared__ int   s_win;

    const int lane = t & 31;
    const int wave = t >> 5;

    if (t == 0) fps_idx[b * kS + 0] = 0;
    float lx = X[0], ly = X[1], lz = X[2];   // first sample = index 0

    for (int s = 1; s < kS; ++s) {
        float bestv = -1.0f;
        int   besti = 0;
#pragma unroll
        for (int j = 0; j < 4; ++j) {
            float dx = px[j] - lx, dy = py[j] - ly, dz = pz[j] - lz;
            float d = dx * dx + dy * dy + dz * dz;
            md[j] = fminf(md[j], d);
            if (md[j] > bestv) { bestv = md[j]; besti = t * 4 + j; }  // j ascending => first index wins
        }
        // wave32 argmax, lowest-index tie-break
        for (int off = 16; off >= 1; off >>= 1) {
            float ov = __shfl_down(bestv, off, 32);
            int   oi = __shfl_down(besti, off, 32);
            if (ov > bestv || (ov == bestv && oi < besti)) { bestv = ov; besti = oi; }
        }
        if (lane == 0) { red_v[wave] = bestv; red_i[wave] = besti; }
        __syncthreads();
        if (wave == 0) {
            float v = red_v[lane];
            int   i = red_i[lane];
            for (int off = 16; off >= 1; off >>= 1) {
                float ov = __shfl_down(v, off, 32);
                int   oi = __shfl_down(i, off, 32);
                if (ov > v || (ov == v && oi < i)) { v = ov; i = oi; }
            }
            if (lane == 0) { s_win = i; fps_idx[b * kS + s] = i; }
        }
        __syncthreads();   // s_win visible; red_* reads (wave0) completed before this
        const int last = s_win;
        lx = X[last * 3 + 0];
        ly = X[last * 3 + 1];
        lz = X[last * 3 + 2];
    }
}

// ---------------------------------------------------------------------------
// Kernel 2: gather fps_xyz (to d_out) and fps_feat (to ws). One block per (b,s).
// ---------------------------------------------------------------------------
__global__ void gather_fps(const float* __restrict__ xyz, const float* __restrict__ feat,
                           const int* __restrict__ fps_idx,
                           float* __restrict__ fps_xyz_out, float* __restrict__ fps_feat_ws)
{
    const int bs = blockIdx.x;
    const int b  = bs / kS;
    const int t  = threadIdx.x;            // 64 threads
    const int idx = fps_idx[bs];
    if (t < 3) fps_xyz_out[(size_t)bs * 3 + t] = xyz[((size_t)b * kN + idx) * 3 + t];
    fps_feat_ws[(size_t)bs * kC + t] = feat[((size_t)b * kN + idx) * kC + t];
}

__global__ void zero_acc(float* acc)
{
    if (threadIdx.x < kB) acc[threadIdx.x] = 0.0f;
}

// ---------------------------------------------------------------------------
// Kernel 3: fused WMMA distance + top-K.
// Block = 512 thr (16 waves) handles 16 queries. Per 256-col chunk each wave
// computes one v_wmma_f32_16x16x4_f32 16x16 tile of D = -2 q.x + |x|^2 into
// LDS (the dropped |q|^2 is a per-row constant -> ordering preserved), then
// each wave scans its own query row maintaining a sorted 32-entry (dist,idx)
// list distributed one per lane (ballot + shuffle insertion, stable:
// ascending dist, lower index first on ties -> matches lax.top_k).
//
// A layout (16x4 f32): v0 = K0 (lanes 0-15) / K2 (16-31); v1 = K1 / K3.
// B layout (4x16 f32): same striping with N on lanes.
// D layout: vgpr r holds (M=r, N=lane) lanes 0-15 and (M=8+r, N=lane-16).
// ---------------------------------------------------------------------------
__global__ __launch_bounds__(512) void topk_kernel(const float* __restrict__ xyz,
                                                   const float* __restrict__ fps_xyz,
                                                   int* __restrict__ group_idx)
{
    const int b    = blockIdx.y;
    const int s0   = blockIdx.x * 16;
    const int wave = threadIdx.x >> 5;
    const int lane = threadIdx.x & 31;

    __shared__ float sq[16 * 256];

    // A fragment: row m = lane&15, (-2qx,-2qy) in lanes<16, (-2qz, 1) in lanes>=16
    const int m = lane & 15;
    const float* Q = fps_xyz + ((size_t)b * kS + s0 + m) * 3;
    const float qx = Q[0], qy = Q[1], qz = Q[2];
    v2f afrag;
    afrag.x = (lane < 16) ? (-2.0f * qx) : (-2.0f * qz);
    afrag.y = (lane < 16) ? (-2.0f * qy) : 1.0f;

    float kd   = 3.0e38f;      // lane's held distance (sorted ascending across lanes)
    int   ki   = 0x7fffffff;   // lane's held index
    float kmax = 3.0e38f;      // cached lane-31 value (refresh only after insertion)

    for (int chunk = 0; chunk < kN; chunk += 256) {
        const int pt = chunk + wave * 16 + (lane & 15);
        const float* P = xyz + ((size_t)b * kN + pt) * 3;
        const float x = P[0], y = P[1], z = P[2];
        const float n2 = x * x + y * y + z * z;
        v2f bfrag;
        bfrag.x = (lane < 16) ? x : z;
        bfrag.y = (lane < 16) ? y : n2;

        v8f cacc = {0.f, 0.f, 0.f, 0.f, 0.f, 0.f, 0.f, 0.f};
        cacc = __builtin_amdgcn_wmma_f32_16x16x4_f32(
            /*neg_a=*/false, afrag, /*neg_b=*/false, bfrag,
            /*c_mod=*/(short)0, cacc, /*reuse_a=*/false, /*reuse_b=*/false);

        // Hide next chunk's point fetch behind the top-k scan phase.
        if (chunk + 256 < kN)
            __builtin_prefetch(xyz + ((size_t)b * kN + pt + 256) * 3, 0, 1);

        const int row0 = (lane < 16) ? 0 : 8;
        const int col  = wave * 16 + (lane & 15);
#pragma unroll
        for (int r = 0; r < 8; ++r) sq[(row0 + r) * 256 + col] = cacc[r];
        __syncthreads();

        const float* rowp = sq + wave * 256;
#pragma unroll 1
        for (int i = 0; i < 256; i += 32) {
            const float cd = rowp[i + lane];
            const int   cj = chunk + i + lane;
            unsigned long long msk = __ballot(cd < kmax);
            while (msk) {
                const int srcl = __ffsll(msk) - 1;   // ascending lane => ascending index (stable)
                msk &= msk - 1;
                const float d = __shfl(cd, srcl, 32);
                const int   j = __shfl(cj, srcl, 32);
                const unsigned long long le = __ballot(kd <= d);
                const int pos = __popcll(le);        // uniform across wave
                if (pos < kK) {
                    const float ud = __shfl_up(kd, 1, 32);
                    const int   ui = __shfl_up(ki, 1, 32);
                    if (lane > pos) { kd = ud; ki = ui; }
                    if (lane == pos) { kd = d; ki = j; }
                    kmax = __shfl(kd, 31, 32);       // uniform branch -> EXEC full
                }
            }
        }
        __syncthreads();
    }
    group_idx[((size_t)b * kS + s0 + wave) * kK + lane] = ki;
}

// ---------------------------------------------------------------------------
// Kernel 4: per-(b,s) group mean over K per channel, and batch-wide sum of
// centered^2 (mean of centered over the whole batch is exactly 0, so
// std = sqrt(sumsq/(n-1))). One wave per (b,s); lane handles 2 channels.
// ---------------------------------------------------------------------------
__global__ __launch_bounds__(128) void stats_kernel(const float* __restrict__ feat,
                                                    const int* __restrict__ group_idx,
                                                    float* __restrict__ mean,
                                                    float* __restrict__ acc)
{
    const int wave = threadIdx.x >> 5;
    const int lane = threadIdx.x & 31;
    const int bs = blockIdx.x * 4 + wave;
    const int b  = bs >> 10;               // kS == 1024
    const int* gi = group_idx + (size_t)bs * kK;
    const int c0 = lane * 2;

    int idxs[kK];
#pragma unroll
    for (int k = 0; k < kK; ++k) idxs[k] = gi[k];

    float s0 = 0.f, s1 = 0.f;
#pragma unroll
    for (int k = 0; k < kK; ++k) {
        const float* fp = feat + ((size_t)b * kN + idxs[k]) * kC + c0;
        s0 += fp[0];
        s1 += fp[1];
    }
    const float m0 = s0 * (1.0f / kK);
    const float m1 = s1 * (1.0f / kK);
    mean[(size_t)bs * kC + c0]     = m0;
    mean[(size_t)bs * kC + c0 + 1] = m1;

    float ss = 0.f;
#pragma unroll
    for (int k = 0; k < kK; ++k) {
        const float* fp = feat + ((size_t)b * kN + idxs[k]) * kC + c0;   // L2-hot re-read
        const float d0 = fp[0] - m0;
        const float d1 = fp[1] - m1;
        ss += d0 * d0 + d1 * d1;
    }
    for (int off = 16; off >= 1; off >>= 1) ss += __shfl_down(ss, off, 32);
    if (lane == 0) atomicAdd(&acc[b], ss);
}

// ---------------------------------------------------------------------------
// Kernel 5: write output (B,S,K,2C): first C = affine-normalized centered
// features, second C = broadcast fps_feat. One block per (b,s).
// ---------------------------------------------------------------------------
__global__ __launch_bounds__(256) void out_kernel(const float* __restrict__ feat,
                                                  const int* __restrict__ group_idx,
                                                  const float* __restrict__ fps_feat,
                                                  const float* __restrict__ mean,
                                                  const float* __restrict__ acc,
                                                  const float* __restrict__ alpha,
                                                  const float* __restrict__ beta,
                                                  float* __restrict__ g_out)
{
    const int bs = blockIdx.x;
    const int b  = bs >> 10;
    const int t  = threadIdx.x;

    __shared__ int   gi_s[kK];
    __shared__ float ff_s[kC], al_s[kC], be_s[kC], mn_s[kC];
    __shared__ float s_inv;

    if (t < kK) gi_s[t] = group_idx[(size_t)bs * kK + t];
    if (t < kC) {
        ff_s[t] = fps_feat[(size_t)bs * kC + t];
        al_s[t] = alpha[t];
        be_s[t] = beta[t];
        mn_s[t] = mean[(size_t)bs * kC + t];
    }
    if (t == 0) {
        const float n1 = (float)((size_t)kS * kK * kC - 1);
        s_inv = 1.0f / (sqrtf(acc[b] / n1) + kEPS);
    }
    __syncthreads();
    const float inv = s_inv;

    float* op = g_out + (size_t)bs * kK * 2 * kC;
#pragma unroll 1
    for (int e = t; e < kK * kC; e += 256) {
        const int k = e >> 6;
        const int c = e & 63;
        const float v = feat[((size_t)b * kN + gi_s[k]) * kC + c];
        const float g = al_s[c] * ((v - mn_s[c]) * inv) + be_s[c];
        op[(size_t)k * 2 * kC + c]      = g;
        op[(size_t)k * 2 * kC + kC + c] = ff_s[c];
    }
}

// ---------------------------------------------------------------------------
// Launch
// ---------------------------------------------------------------------------
extern "C" void kernel_launch(void* const* d_in, const int* in_sizes, int n_in,
                              void* d_out, int out_size, void* d_ws, size_t ws_size,
                              hipStream_t stream)
{
    const float* xyz   = (const float*)d_in[0];   // (B,N,3)
    const float* feat  = (const float*)d_in[1];   // (B,N,C)
    const float* alpha = (const float*)d_in[2];   // (1,1,1,C)
    const float* beta  = (const float*)d_in[3];   // (1,1,1,C)
    float* out = (float*)d_out;

    // Workspace layout (all 4B aligned)
    char* ws = (char*)d_ws;
    size_t off = 0;
    int*   fps_idx   = (int*)(ws + off);   off += (size_t)kB * kS * sizeof(int);        // 32 KB
    int*   group_idx = (int*)(ws + off);   off += (size_t)kB * kS * kK * sizeof(int);   // 1 MB
    float* fps_feat  = (float*)(ws + off); off += (size_t)kB * kS * kC * sizeof(float); // 2 MB
    float* mean      = (float*)(ws + off); off += (size_t)kB * kS * kC * sizeof(float); // 2 MB
    float* acc       = (float*)(ws + off); off += (size_t)kB * sizeof(float);

    float* fps_xyz_out = out;                         // (B,S,3)
    float* g_out       = out + (size_t)kB * kS * 3;   // (B,S,K,2C)

    fps_kernel<<<kB, 1024, 0, stream>>>(xyz, fps_idx);
    gather_fps<<<kB * kS, 64, 0, stream>>>(xyz, feat, fps_idx, fps_xyz_out, fps_feat);
    zero_acc<<<1, 32, 0, stream>>>(acc);
    topk_kernel<<<dim3(kS / 16, kB), 512, 0, stream>>>(xyz, fps_xyz_out, group_idx);
    stats_kernel<<<kB * kS / 4, 128, 0, stream>>>(feat, group_idx, mean, acc);
    out_kernel<<<kB * kS, 256, 0, stream>>>(feat, group_idx, fps_feat, mean, acc,
                                            alpha, beta, g_out);
}